// HSCD_37864431682565
// MI455X (gfx1250) — compile-verified
//
#include <hip/hip_runtime.h>
#include <math.h>

// ---------------------------------------------------------------------------
// HSCD: 5x GCN-propagate (degree-norm gather/scatter + row L2 normalize +
// residual) then mean-fuse. Memory/atomic bound (AI ~0.2 FLOP/B) -> no WMMA;
// optimize for 23.3 TB/s HBM + 192MB L2 + wave32. CDNA5-specific path used:
// async global->LDS staging (ASYNCcnt) of the COO edge stream, double-buffered
// per block, plus global_prefetch_b8 on the random gather rows.
// ---------------------------------------------------------------------------

#define DDIM   64
#define DDIM4  16          // float4 per node row
#define NUROWS 150001
#define NNODES 230002
#define BS     256
#define EPB    1024        // edges staged per chunk (4KB src + 4KB dst)
#define NCH    4           // chunks per block (double-buffered)

#define AS1 __attribute__((address_space(1)))
#define AS3 __attribute__((address_space(3)))

typedef int v4i_ __attribute__((vector_size(16)));   // matches builtin param

// ---- CDNA5 async global->LDS (ASYNCcnt) helpers ---------------------------
__device__ __forceinline__ void async_ld_b128(void* lds, const void* g) {
#if __has_builtin(__builtin_amdgcn_global_load_async_to_lds_b128)
  __builtin_amdgcn_global_load_async_to_lds_b128(
      (AS1 v4i_*)g, (AS3 v4i_*)lds, 0, 0);
#else
  unsigned lofs = (unsigned)(size_t)(AS3 void*)lds;   // 32-bit LDS byte addr
  asm volatile("global_load_async_to_lds_b128 %0, %1, off"
               :: "v"(lofs), "v"(g) : "memory");
#endif
}

__device__ __forceinline__ void wait_async_0() {
#if __has_builtin(__builtin_amdgcn_s_wait_asynccnt)
  __builtin_amdgcn_s_wait_asynccnt(0);
#else
  asm volatile("s_wait_asynccnt 0x0" ::: "memory");
#endif
}

__device__ __forceinline__ void wait_async_2() {
#if __has_builtin(__builtin_amdgcn_s_wait_asynccnt)
  __builtin_amdgcn_s_wait_asynccnt(2);
#else
  asm volatile("s_wait_asynccnt 0x2" ::: "memory");
#endif
}

// ---------------------------------------------------------------------------

__global__ __launch_bounds__(BS) void zero_f(float* __restrict__ p, int n) {
  int i = blockIdx.x * blockDim.x + threadIdx.x;
  if (i < n) p[i] = 0.0f;
}

// x0 = concat(user_table, item_table) along rows, as flat float4 stream.
__global__ __launch_bounds__(BS) void concat_x0(const float4* __restrict__ u,
                                                const float4* __restrict__ it,
                                                float4* __restrict__ x0,
                                                int nu4, int ntot4) {
  int i = blockIdx.x * blockDim.x + threadIdx.x;
  if (i < ntot4) x0[i] = (i < nu4) ? u[i] : it[i - nu4];
}

// deg[dst] += 1 over the edge list (L2-side f32 atomics).
__global__ __launch_bounds__(BS) void degree_k(const int* __restrict__ dst, int E,
                                               float* __restrict__ deg) {
  int e = blockIdx.x * blockDim.x + threadIdx.x;
  if (e < E) atomicAdd(&deg[dst[e]], 1.0f);
}

// In-place deg -> dis = (deg>0) ? rsqrt(max(deg,1)) : 0
__global__ __launch_bounds__(BS) void dis_k(float* __restrict__ deg, int n) {
  int i = blockIdx.x * blockDim.x + threadIdx.x;
  if (i < n) {
    float v = deg[i];
    deg[i] = (v > 0.0f) ? rsqrtf(fmaxf(v, 1.0f)) : 0.0f;
  }
}

// Main scatter: h[dst] += x[src] * dis[src].
// Edge indices double-buffered global->LDS via async engine (ASYNCcnt);
// 16 lanes per edge, float4 per lane; prefetch gather rows 128 edges ahead.
__global__ __launch_bounds__(BS) void scatter_async_k(const float4* __restrict__ x4,
                                                      const float* __restrict__ dis,
                                                      const int* __restrict__ src,
                                                      const int* __restrict__ dst,
                                                      int nChunks,
                                                      float* __restrict__ h) {
  __shared__ int sIdx[2][2][EPB];          // [buf][0=src,1=dst][edge]
  const int t = threadIdx.x;
  const int base = blockIdx.x * NCH;

  // Stage first chunk into buffer 0 (2 async b128 ops per wave).
  async_ld_b128(&sIdx[0][0][t * 4], src + (size_t)base * EPB + t * 4);
  async_ld_b128(&sIdx[0][1][t * 4], dst + (size_t)base * EPB + t * 4);

  for (int c = 0; c < NCH; ++c) {
    const int chunk = base + c;
    if (chunk >= nChunks) break;

    const bool haveNext = (c + 1 < NCH) && (chunk + 1 < nChunks);
    if (haveNext) {
      const int nb = (c + 1) & 1;
      async_ld_b128(&sIdx[nb][0][t * 4], src + (size_t)(chunk + 1) * EPB + t * 4);
      async_ld_b128(&sIdx[nb][1][t * 4], dst + (size_t)(chunk + 1) * EPB + t * 4);
      wait_async_2();                      // current chunk's 2 ops retired
    } else {
      wait_async_0();
    }
    __syncthreads();                       // all waves' staging visible

    const int buf = c & 1;
    const int g = t >> 4;                  // edge subgroup 0..15
    const int j = t & 15;                  // float4 slot within row
#pragma unroll 8
    for (int p = 0; p < EPB / 16; ++p) {
      const int ei = p * 16 + g;
      const int s = sIdx[buf][0][ei];
      const int d = sIdx[buf][1][ei];
      if (j == 0) {
        const int pp = ei + 128;           // look-ahead on random gather rows
        if (pp < EPB)
          __builtin_prefetch(&x4[(size_t)sIdx[buf][0][pp] * DDIM4], 0, 3);
      }
      const float sc = dis[s];
      const float4 v = x4[(size_t)s * DDIM4 + j];
      float* hp = h + (size_t)d * DDIM + (size_t)j * 4;
      atomicAdd(hp + 0, v.x * sc);
      atomicAdd(hp + 1, v.y * sc);
      atomicAdd(hp + 2, v.z * sc);
      atomicAdd(hp + 3, v.w * sc);
    }
    __syncthreads();                       // buffer safe to restage
  }
}

// Tail scatter for E mod EPB edges (plain path).
__global__ __launch_bounds__(BS) void scatter_k(const float4* __restrict__ x4,
                                                const float* __restrict__ dis,
                                                const int* __restrict__ src,
                                                const int* __restrict__ dst,
                                                int E, float* __restrict__ h) {
  int t = blockIdx.x * blockDim.x + threadIdx.x;
  int e = t >> 4;
  int j = t & 15;
  if (e >= E) return;
  int s = src[e];
  int d = dst[e];
  float sc = dis[s];
  float4 v = x4[(size_t)s * DDIM4 + j];
  float* hp = h + (size_t)d * DDIM + (size_t)j * 4;
  atomicAdd(hp + 0, v.x * sc);
  atomicAdd(hp + 1, v.y * sc);
  atomicAdd(hp + 2, v.z * sc);
  atomicAdd(hp + 3, v.w * sc);
}

// Fused: hn = h*dis[node]; r = x + hn/max(||hn||,1e-12); h <- r; acc += r.
// 16 lanes per node (float4 each); xor-shuffle reduce stays in the half-wave.
__global__ __launch_bounds__(BS) void finalize_k(const float4* __restrict__ x4,
                                                 float4* __restrict__ h4,
                                                 const float* __restrict__ dis,
                                                 float4* __restrict__ acc4, int n) {
  int t = blockIdx.x * blockDim.x + threadIdx.x;
  int node = t >> 4;
  int j = t & 15;
  if (node >= n) return;

  size_t idx = (size_t)node * DDIM4 + j;
  float dd = dis[node];
  float4 hv = h4[idx];
  hv.x *= dd; hv.y *= dd; hv.z *= dd; hv.w *= dd;

  float ss = hv.x * hv.x + hv.y * hv.y + hv.z * hv.z + hv.w * hv.w;
  ss += __shfl_xor(ss, 1);
  ss += __shfl_xor(ss, 2);
  ss += __shfl_xor(ss, 4);
  ss += __shfl_xor(ss, 8);                 // masks <16: never crosses nodes

  float inv = 1.0f / fmaxf(sqrtf(ss), 1e-12f);

  float4 xv = x4[idx];
  float4 r;
  r.x = xv.x + hv.x * inv;
  r.y = xv.y + hv.y * inv;
  r.z = xv.z + hv.z * inv;
  r.w = xv.w + hv.w * inv;
  h4[idx] = r;                             // behavior embedding (reused as x later)

  float4 a = acc4[idx];
  a.x += r.x; a.y += r.y; a.z += r.z; a.w += r.w;
  acc4[idx] = a;                           // running fusion sum
}

__global__ __launch_bounds__(BS) void scale_k(float4* __restrict__ p, int n4, float s) {
  int i = blockIdx.x * blockDim.x + threadIdx.x;
  if (i < n4) {
    float4 v = p[i];
    v.x *= s; v.y *= s; v.z *= s; v.w *= s;
    p[i] = v;
  }
}

extern "C" void kernel_launch(void* const* d_in, const int* in_sizes, int n_in,
                              void* d_out, int out_size, void* d_ws, size_t ws_size,
                              hipStream_t stream) {
  (void)n_in; (void)out_size; (void)ws_size;

  const float* user = (const float*)d_in[0];
  const float* item = (const float*)d_in[1];
  const int* e_ubg  = (const int*)d_in[2];  int E0 = in_sizes[2] / 2;
  const int* e_view = (const int*)d_in[3];  int E1 = in_sizes[3] / 2;
  const int* e_cart = (const int*)d_in[4];  int E2 = in_sizes[4] / 2;
  const int* e_vb   = (const int*)d_in[5];  int E3 = in_sizes[5] / 2;
  const int* e_cb   = (const int*)d_in[6];  int E4 = in_sizes[6] / 2;

  float* ws = (float*)d_ws;
  const size_t NPAD = ((size_t)NNODES + 63) & ~(size_t)63;
  const long long ND = (long long)NNODES * DDIM;      // 14,720,128 floats

  float* deg = ws;                 // N floats (reused per layer: deg -> dis)
  float* A   = ws + NPAD;          // N x D
  float* B   = A + (size_t)ND;     // N x D
  float* C   = B + (size_t)ND;     // N x D
  float* out = (float*)d_out;      // N x D fusion accumulator

  auto cdiv = [](long long a, long long b) { return (int)((a + b - 1) / b); };

  // Fusion accumulator starts at zero (d_ws/d_out arrive poisoned).
  zero_f<<<cdiv(ND, BS), BS, 0, stream>>>(out, (int)ND);

  // x0 = concat(user, item)  -> A
  concat_x0<<<cdiv(ND / 4, BS), BS, 0, stream>>>(
      (const float4*)user, (const float4*)item, (float4*)A,
      NUROWS * DDIM4, (int)(ND / 4));

  auto prop = [&](const float* x, const int* e, int E, float* h) {
    const int* src = e;
    const int* dst = e + E;
    zero_f<<<cdiv((long long)NNODES, BS), BS, 0, stream>>>(deg, NNODES);
    zero_f<<<cdiv(ND, BS), BS, 0, stream>>>(h, (int)ND);
    degree_k<<<cdiv((long long)E, BS), BS, 0, stream>>>(dst, E, deg);
    dis_k<<<cdiv((long long)NNODES, BS), BS, 0, stream>>>(deg, NNODES);

    const int nChunks = E / EPB;
    if (nChunks > 0) {
      scatter_async_k<<<cdiv(nChunks, NCH), BS, 0, stream>>>(
          (const float4*)x, deg, src, dst, nChunks, h);
    }
    const int erem = E - nChunks * EPB;
    if (erem > 0) {
      scatter_k<<<cdiv((long long)erem * 16, BS), BS, 0, stream>>>(
          (const float4*)x, deg, src + (size_t)nChunks * EPB,
          dst + (size_t)nChunks * EPB, erem, h);
    }
    finalize_k<<<cdiv((long long)NNODES * 16, BS), BS, 0, stream>>>(
        (const float4*)x, (float4*)h, deg, (float4*)out, NNODES);
  };

  prop(A, e_ubg,  E0, B);   // B = ubg        (x0 in A now dead)
  prop(B, e_view, E1, C);   // C = view
  prop(B, e_cart, E2, A);   // A = cart       (ubg consumed after this)
  prop(C, e_vb,   E3, B);   // B = view_buy   (overwrites dead ubg)
  prop(A, e_cb,   E4, C);   // C = cart_buy   (overwrites dead view)

  // final = sum / 5
  scale_k<<<cdiv(ND / 4, BS), BS, 0, stream>>>((float4*)out, (int)(ND / 4), 0.2f);
}